// Net_56392920596955
// MI455X (gfx1250) — compile-verified
//
#include <hip/hip_runtime.h>
#include <hip/hip_bf16.h>

// Persistent single-workgroup FDTD wave propagator for MI455X (gfx1250).
// Laplacian computed as banded matrix products D*P + P*D via
// V_WMMA_F32_16X16X4_F32 (fp32 matrix pipe, exact reference precision).
// All tile-level control flow is wave-uniform and forced through SGPRs
// (readfirstlane) so WMMA always executes with EXEC = all ones.

typedef float v8f __attribute__((ext_vector_type(8)));
typedef float v2f __attribute__((ext_vector_type(2)));

#define NXg   256
#define NZg   256
#define NTS   1000
#define NRECn 200
#define NBw   40
#define ALPHAc 0.0053f
#define DTc    0.001f
#define NPTS  (NXg * NZg)

__device__ __forceinline__ v8f wmma4(v2f a, v2f b, v8f c) {
  // 8 args: (neg_a, A, neg_b, B, c_mod, C, reuse_a, reuse_b)
  return __builtin_amdgcn_wmma_f32_16x16x4_f32(false, a, false, b, (short)0, c,
                                               false, false);
}

__device__ __forceinline__ float taper1(int i) {
  int d = (i < NXg - 1 - i) ? i : (NXg - 1 - i);
  if (d < NBw) {
    float a = ALPHAc * (float)(NBw - d);
    return expf(-a * a);
  }
  return 1.0f;
}

// banded second-difference value: band(r,c) = coeff[|c-r|], zero beyond band 2
__device__ __forceinline__ float bandv(int r, int c, float cA, float cB, float cC) {
  int d = c - r; d = d < 0 ? -d : d;
  return d == 0 ? cC : d == 1 ? cB : d == 2 ? cA : 0.0f;
}

__global__ __launch_bounds__(1024) void fdtd_wmma(
    const float* __restrict__ q,     // [1000] ricker wavelet
    const float* __restrict__ lw,    // [25] 5x5 laplace kernel
    const float* __restrict__ m,     // [256*256] slowness^2
    const int*   __restrict__ xrec,  // [200]
    const int*   __restrict__ zrec,  // [200]
    const int*   __restrict__ sxp,   // [1]
    const int*   __restrict__ szp,   // [1]
    float*       __restrict__ out,   // [1000*200 + 256*256]
    float*       __restrict__ ws)    // >= 4*NPTS floats
{
  float* buf0 = ws;                  // pressure ping
  float* buf1 = ws + NPTS;           // pressure pong
  float* dc   = ws + 2 * NPTS;       // interleaved {damp, dt^2/m} pairs

  const int tid  = threadIdx.x;                              // 0..1023
  const int lane = tid & 31;                                 // wave32 lane
  const int wid  = __builtin_amdgcn_readfirstlane(tid >> 5); // SGPR wave id
  const int half = lane >> 4;                                // 0|1 lane-half
  const int lm   = lane & 15;

  // ---- init phase: damping mask, dt^2/m, zero fields ----
  for (int i = tid; i < NPTS; i += 1024) {
    int x = i >> 8, z = i & 255;
    dc[2 * i]     = taper1(x) * taper1(z);
    dc[2 * i + 1] = (DTc * DTc) / m[i];
    buf0[i] = 0.0f;
    buf1[i] = 0.0f;
  }

  const int sx = sxp[0];
  const int sz = szp[0];

  // 1-D stencil coeffs from the conv weight, row 2: [cA cB 2*cC cB cA]
  const float cA = lw[10];
  const float cB = lw[11];
  const float cC = 0.5f * lw[12];

  // ---- constant WMMA operands (built once, per lane) ----
  // A-layout f32 16x4: lane holds (m=lm, k=v+2*half); B-layout 4x16: (k=v+2*half, n=lm)
  v2f Mdiag[4];                      // banded diag block; symmetric => valid as A or B
#pragma unroll
  for (int s = 0; s < 4; ++s) {
    int k0 = 4 * s + 2 * half;
    Mdiag[s].x = bandv(lm, k0,     cA, cB, cC);
    Mdiag[s].y = bandv(lm, k0 + 1, cA, cB, cC);
  }
  v2f Aup, Adn, Bleft, Bright;
  {
    int ku = 12 + 2 * half;
    int kd = 2 * half;
#pragma unroll
    for (int v = 0; v < 2; ++v) {
      int k1 = ku + v, k2 = kd + v;
      // x-dir upper-neighbor A block: (m,k) = (0,14)cA (0,15)cB (1,15)cA, K-slice 12..15
      float au = (lm == 0 && k1 == 14) ? cA : (lm == 0 && k1 == 15) ? cB
               : (lm == 1 && k1 == 15) ? cA : 0.0f;
      // x-dir lower-neighbor A block: (14,0)cA (15,0)cB (15,1)cA, K-slice 0..3
      float ad = (lm == 14 && k2 == 0) ? cA : (lm == 15 && k2 == 0) ? cB
               : (lm == 15 && k2 == 1) ? cA : 0.0f;
      // z-dir left-neighbor B block: (k,n) = (14,0)cA (15,0)cB (15,1)cA, K-slice 12..15
      float bl = (k1 == 14 && lm == 0) ? cA : (k1 == 15 && lm == 0) ? cB
               : (k1 == 15 && lm == 1) ? cA : 0.0f;
      // z-dir right-neighbor B block: (0,14)cA (0,15)cB (1,15)cA, K-slice 0..3
      float br = (k2 == 0 && lm == 14) ? cA : (k2 == 0 && lm == 15) ? cB
               : (k2 == 1 && lm == 15) ? cA : 0.0f;
      if (v == 0) { Aup.x = au; Adn.x = ad; Bleft.x = bl; Bright.x = br; }
      else        { Aup.y = au; Adn.y = ad; Bleft.y = bl; Bright.y = br; }
    }
  }

  __threadfence_block();
  __syncthreads();

  // ---- time loop: one barrier per step ----
#pragma unroll 1
  for (int t = 0; t < NTS; ++t) {
    const float* cur = (t & 1) ? buf0 : buf1;  // p_curr
    float*       nxt = (t & 1) ? buf1 : buf0;  // holds p_prev, overwritten with p_new
    const float  qt  = q[t];

#pragma unroll 1
    for (int j = 0; j < 8; ++j) {              // 32 waves x 8 tiles = 256 tiles
      // all-scalar tile coordinates -> uniform s_cbranch control flow
      const int tile = wid * 8 + j;
      const int bi = tile >> 4, bj = tile & 15;
      const int xb = bi << 4, zb = bj << 4;

      v8f acc = {0.f, 0.f, 0.f, 0.f, 0.f, 0.f, 0.f, 0.f};

      // ---- x-direction: D @ P (A = const D blocks, B = P row-slices) ----
      if (bi > 0) {
        int r0 = ((bi - 1) << 4) + 12 + 2 * half;
        v2f b; b.x = cur[r0 * NZg + zb + lm]; b.y = cur[(r0 + 1) * NZg + zb + lm];
        acc = wmma4(Aup, b, acc);
      }
#pragma unroll
      for (int s = 0; s < 4; ++s) {
        int r0 = xb + 4 * s + 2 * half;
        v2f b; b.x = cur[r0 * NZg + zb + lm]; b.y = cur[(r0 + 1) * NZg + zb + lm];
        acc = wmma4(Mdiag[s], b, acc);
      }
      if (bi < 15) {
        int r0 = ((bi + 1) << 4) + 2 * half;
        v2f b; b.x = cur[r0 * NZg + zb + lm]; b.y = cur[(r0 + 1) * NZg + zb + lm];
        acc = wmma4(Adn, b, acc);
      }

      // ---- z-direction: P @ D (A = P col-slices as aligned b64, B = const D) ----
      {
        const float* rowp = cur + (xb + lm) * NZg;
        if (bj > 0) {
          int c0 = ((bj - 1) << 4) + 12 + 2 * half;          // even -> 8B aligned
          v2f a = *(const v2f*)(rowp + c0);
          acc = wmma4(a, Bleft, acc);
        }
#pragma unroll
        for (int s = 0; s < 4; ++s) {
          int c0 = zb + 4 * s + 2 * half;                    // even -> 8B aligned
          v2f a = *(const v2f*)(rowp + c0);
          acc = wmma4(a, Mdiag[s], acc);
        }
        if (bj < 15) {
          int c0 = ((bj + 1) << 4) + 2 * half;               // even -> 8B aligned
          v2f a = *(const v2f*)(rowp + c0);
          acc = wmma4(a, Bright, acc);
        }
      }

      // ---- pointwise update in C/D layout; fold in source injection ----
#pragma unroll
      for (int g = 0; g < 8; ++g) {
        int row = xb + g + 8 * half;
        int col = zb + lm;
        int idx = row * NZg + col;
        float pc = cur[idx];
        float pp = nxt[idx];                       // p_prev (same thread overwrites)
        v2f  d2  = *(const v2f*)(dc + 2 * idx);    // {damp, dt^2/m} one b64 load
        float dm = d2.x, cf = d2.y;
        float pn = dm * (cf * acc[g] + 2.0f * pc - dm * pp);
        if (row == sx && col == sz) pn += dm * qt;
        nxt[idx] = pn;
      }
    }

    __threadfence_block();
    __syncthreads();
    __threadfence_block();

    // receiver gather (post-injection field)
    if (tid < NRECn) {
      out[t * NRECn + tid] = nxt[xrec[tid] * NZg + zrec[tid]];
    }
  }

  __threadfence_block();
  __syncthreads();

  // final field: written at t=999 (odd) -> buf1
  for (int i = tid; i < NPTS; i += 1024) {
    out[NTS * NRECn + i] = buf1[i];
  }
}

extern "C" void kernel_launch(void* const* d_in, const int* in_sizes, int n_in,
                              void* d_out, int out_size, void* d_ws, size_t ws_size,
                              hipStream_t stream) {
  const float* q    = (const float*)d_in[0];
  const float* lw   = (const float*)d_in[1];
  const float* m    = (const float*)d_in[2];
  const int*   xrec = (const int*)d_in[3];
  const int*   zrec = (const int*)d_in[4];
  const int*   sxp  = (const int*)d_in[5];
  const int*   szp  = (const int*)d_in[6];
  float* out = (float*)d_out;
  float* ws  = (float*)d_ws;   // needs 4*256*256 floats = 1 MB

  hipLaunchKernelGGL(fdtd_wmma, dim3(1), dim3(1024), 0, stream,
                     q, lw, m, xrec, zrec, sxp, szp, out, ws);
}